// ArDCA_84920093377316
// MI455X (gfx1250) — compile-verified
//
#include <hip/hip_runtime.h>

typedef float v2f __attribute__((ext_vector_type(2)));
typedef float v8f __attribute__((ext_vector_type(8)));
typedef unsigned int u32x4 __attribute__((ext_vector_type(4)));
typedef int i32x8 __attribute__((ext_vector_type(8)));
typedef int i32x4 __attribute__((ext_vector_type(4)));

#define M_DIM 512
#define L_DIM 256
#define Q_DIM 21
#define XROW  (L_DIM * Q_DIM)            // 5376 floats: X row / out row stride
#define JISTR (L_DIM * Q_DIM * Q_DIM)    // 112896 floats: J stride per i
#define BK    128                        // K-rows per LDS chunk
#define CHNK  (BK * Q_DIM)               // 2688 dwords per chunk
#define BPAD  16                         // tail pad so fragment reads stay in-bounds

#if __has_builtin(__builtin_amdgcn_tensor_load_to_lds)
#define USE_TDM 1
#else
#define USE_TDM 0
#endif

#if USE_TDM
// 1-D TDM copy: n_tile dwords requested, reads past n_valid dwords return 0.
__device__ __forceinline__ void tdm_load_1d(unsigned lds_byte_off,
                                            const float* gptr,
                                            unsigned n_valid,   // dwords in tensor
                                            unsigned n_tile)    // dwords in tile
{
    unsigned long long ga = (unsigned long long)(uintptr_t)gptr;

    u32x4 g0;
    g0[0] = 1u;                                   // count=1, user descriptor
    g0[1] = lds_byte_off;                         // lds_addr
    g0[2] = (unsigned)ga;                         // global_addr[31:0]
    g0[3] = (unsigned)((ga >> 32) & 0x01FFFFFFu)  // global_addr[56:32]
          | (2u << 30);                           // type=2 ("image")

    i32x8 g1;
    g1[0] = (int)(2u << 16);                      // data_size=2 (4 bytes); mask/flags 0
    g1[1] = (int)((n_valid & 0xFFFFu) << 16);     // tensor_dim0[15:0]  @ bits 63:48
    g1[2] = (int)((n_valid >> 16) | (1u << 16));  // tensor_dim0[31:16]; tensor_dim1=1
    g1[3] = (int)((n_tile & 0xFFFFu) << 16);      // tile_dim0 @ bits 127:112
    g1[4] = 0;                                    // tile_dim1=0, tile_dim2=0 (unused)
    g1[5] = (int)n_valid;                         // tensor_dim0_stride[31:0] (benign for 1-D)
    g1[6] = 0;
    g1[7] = 0;

    i32x4 z4 = {0, 0, 0, 0};                      // groups 2/3: dims unused
    i32x8 z8 = {0, 0, 0, 0, 0, 0, 0, 0};          // trailing group (6-arg toolchain form)
    __builtin_amdgcn_tensor_load_to_lds(g0, g1, z4, z4, z8, 0);
}
#endif

__global__ __launch_bounds__(256)
void ardca_wmma_f32(const float* __restrict__ X,
                    const float* __restrict__ h,
                    const float* __restrict__ J,
                    float* __restrict__ out)
{
    __shared__ float B0[CHNK + BPAD];
    __shared__ float B1[CHNK + BPAD];

    const int i    = blockIdx.x;          // position index; K-depth = i*Q
    const int K    = i * Q_DIM;
    const int tid  = threadIdx.x;
    const int lane = tid & 31;
    const int wave = tid >> 5;
    const int lo   = lane & 15;
    const int hi   = lane >> 4;
    const int m0   = blockIdx.y * 128 + wave * 16;

    const float* __restrict__ Jb   = J + (size_t)i * JISTR;        // B (K x 21), row-major, contiguous
    const float* __restrict__ Arow = X + (size_t)(m0 + lo) * XROW; // this lane's A row

    // zero the pad tails once (keeps stray fragment reads finite/zero)
    if (tid < BPAD) { B0[CHNK + tid] = 0.0f; B1[CHNK + tid] = 0.0f; }

    v8f acc0 = {};                        // C cols 0..15
    v8f acc1 = {};                        // C cols 16..20 (+discarded pad cols)

    const int nch = (K + BK - 1) / BK;

#if USE_TDM
    if (wave == 0 && nch > 0)
        tdm_load_1d((unsigned)(size_t)(void*)B0, Jb, (unsigned)(K * Q_DIM), CHNK);

    for (int c = 0; c < nch; ++c) {
        const int k0 = c * BK;
        float* Bc = (c & 1) ? B1 : B0;
        float* Bn = (c & 1) ? B0 : B1;

        __builtin_amdgcn_s_wait_tensorcnt(0);   // chunk c landed in LDS (issuing wave)
        __syncthreads();                        // publish chunk c; chunk c-1 buffer now free
        if (wave == 0 && c + 1 < nch)           // overlap DMA of c+1 with compute of c
            tdm_load_1d((unsigned)(size_t)(void*)Bn,
                        Jb + (size_t)(k0 + BK) * Q_DIM,
                        (unsigned)((K - (k0 + BK)) * Q_DIM), CHNK);
#else
    for (int c = 0; c < nch; ++c) {
        const int k0 = c * BK;
        float* Bc = B0;
        const int lim = (K - k0) * Q_DIM;       // valid dwords in this chunk
        __syncthreads();                        // previous chunk fully consumed
        for (int e = tid; e < CHNK; e += 256)
            Bc[e] = (e < lim) ? Jb[(size_t)k0 * Q_DIM + e] : 0.0f;
        if (k0 + BK < K)
            __builtin_prefetch(Jb + (size_t)(k0 + BK) * Q_DIM, 0, 1);
        __syncthreads();
#endif

        // --- 32 K-steps of 4: v_wmma_f32_16x16x4_f32 ---
        #pragma unroll
        for (int kk = 0; kk < BK; kk += 4) {
            // A: lane holds A[m0+lo][kb..kb+1]; kb even -> 8B aligned; k>=K cols hit zeroed B rows
            const int kb = k0 + kk + 2 * hi;
            v2f a = *(const v2f*)(Arow + kb);

            // B (packed stride-21): element (K = kk+2*hi+v, N = n0+lo)
            const int rb = (kk + 2 * hi) * Q_DIM;
            v2f b0 = { Bc[rb + lo],          Bc[rb + Q_DIM + lo] };
            v2f b1 = { Bc[rb + 16 + lo],     Bc[rb + Q_DIM + 16 + lo] };

            acc0 = __builtin_amdgcn_wmma_f32_16x16x4_f32(
                       false, a, false, b0, (short)0, acc0, false, false);
            acc1 = __builtin_amdgcn_wmma_f32_16x16x4_f32(
                       false, a, false, b1, (short)0, acc1, false, false);
        }
    }

    // --- epilogue: C reg r -> row m0 + r + 8*hi, col n0 + lo ---
    const float hv0 = h[i * Q_DIM + lo];
    const float hv1 = (lo < 5) ? h[i * Q_DIM + 16 + lo] : 0.0f;
    #pragma unroll
    for (int r = 0; r < 8; ++r) {
        const int m = m0 + r + 8 * hi;
        float* orow = out + (size_t)m * XROW + i * Q_DIM;
        orow[lo] = acc0[r] + hv0;
        if (lo < 5) orow[16 + lo] = acc1[r] + hv1;
    }
}

extern "C" void kernel_launch(void* const* d_in, const int* in_sizes, int n_in,
                              void* d_out, int out_size, void* d_ws, size_t ws_size,
                              hipStream_t stream)
{
    const float* X = (const float*)d_in[0];   // X_oh: (512, 256, 21) f32
    const float* h = (const float*)d_in[1];   // h_pos: (256, 21) f32
    const float* J = (const float*)d_in[2];   // J: (256, 256, 21, 21) f32
    float* out = (float*)d_out;               // (512, 256, 21) f32

    dim3 grid(L_DIM, M_DIM / 128);            // 256 x 4 blocks
    ardca_wmma_f32<<<grid, 256, 0, stream>>>(X, h, J, out);
}